// LSTMModel_48180943126535
// MI455X (gfx1250) — compile-verified
//
#include <hip/hip_runtime.h>
#include <hip/hip_bf16.h>
#include <math.h>

// ---------------------------------------------------------------------------
// Model dims (fixed by the reference)
// ---------------------------------------------------------------------------
#define BB   128          // batch
#define TT   256          // time steps
#define FCGM 32
#define FOTH 64
#define HH   512          // hidden
#define G4H  2048         // 4*H gate width
#define NHEAD 8
#define HD   64
#define DD1  256
#define DD2  128
#define ROWS (BB*TT)      // 32768 flattened rows

typedef __attribute__((ext_vector_type(2))) float v2f;
typedef __attribute__((ext_vector_type(8))) float v8f;

__device__ __forceinline__ v8f wmma4(v2f a, v2f b, v8f c) {
  // D = A(16x4 f32) * B(4x16 f32) + C(16x16 f32)
  return __builtin_amdgcn_wmma_f32_16x16x4_f32(false, a, false, b,
                                               (short)0, c, false, false);
}

__device__ __forceinline__ float sigmf(float x) {
  return 1.0f / (1.0f + __expf(-x));
}

// CDNA5 async global->LDS copy, 16 bytes per lane, tracked by ASYNCcnt.
// lds_off is the 32-bit LDS byte address (low 32 bits of a flat shared ptr).
__device__ __forceinline__ void async_copy16(unsigned lds_off,
                                             const float* g) {
  asm volatile("global_load_async_to_lds_b128 %0, %1, off"
               :: "v"(lds_off), "v"(g) : "memory");
}
__device__ __forceinline__ void wait_async0() {
  asm volatile("s_wait_asynccnt 0" ::: "memory");
}

// ---------------------------------------------------------------------------
// Generic WMMA f32 GEMM: C[M,N] = act(A[M,K] @ W[K,N] + bias[N])
// Requirements: M%64==0, N%64==0, K%16==0, lda%4==0, ldw%4==0 (16B-aligned
// rows for the async b128 staging).  Block = 128 thr (4 waves), block tile
// 64x64, wave tile 16x64 (4 accumulators of 16x16).
// act: 0=none, 1=relu, 2=sigmoid
// ---------------------------------------------------------------------------
__global__ __launch_bounds__(128)
void k_gemm(const float* __restrict__ A, int lda,
            const float* __restrict__ W, int ldw,
            const float* __restrict__ bias,
            float* __restrict__ Co, int ldc,
            int M, int N, int K, int act)
{
  __shared__ float As[64][20];   // stride 20: 16B-aligned rows, conflict-free
  __shared__ float Ws[16][64];

  const int tid  = threadIdx.x;
  const int lane = tid & 31;
  const int wave = tid >> 5;       // 0..3 -> M sub-tile
  const int half = lane >> 4;      // 0/1
  const int lm   = lane & 15;
  const int nb   = blockIdx.x * 64;
  const int mb   = blockIdx.y * 64;

  const unsigned As_base = (unsigned)(size_t)&As[0][0];
  const unsigned Ws_base = (unsigned)(size_t)&Ws[0][0];

  v8f acc[4];
  #pragma unroll
  for (int nt = 0; nt < 4; ++nt) {
    float bv = bias ? bias[nb + nt * 16 + lm] : 0.0f;
    v8f a0 = {bv, bv, bv, bv, bv, bv, bv, bv};
    acc[nt] = a0;
  }

  #pragma unroll 1
  for (int kt = 0; kt < K; kt += 16) {
    // ---- async-stage A (64x16) and W (16x64) tiles into LDS ----
    {
      const float* Ab = A + (size_t)mb * lda + kt;
      #pragma unroll
      for (int i = 0; i < 2; ++i) {          // 256 chunks of 16B
        int cidx = tid + i * 128;
        int r = cidx >> 2, c4 = (cidx & 3) << 2;
        async_copy16(As_base + (unsigned)((r * 20 + c4) * 4),
                     Ab + (size_t)r * lda + c4);
      }
      const float* Wb = W + (size_t)kt * ldw + nb;
      #pragma unroll
      for (int i = 0; i < 2; ++i) {          // 256 chunks of 16B
        int cidx = tid + i * 128;
        int r = cidx >> 4, c4 = (cidx & 15) << 2;
        async_copy16(Ws_base + (unsigned)((r * 64 + c4) * 4),
                     Wb + (size_t)r * ldw + c4);
      }
      wait_async0();
    }
    __syncthreads();

    #pragma unroll
    for (int kk = 0; kk < 4; ++kk) {
      const int k0 = kk * 4 + half * 2;                       // even
      v2f a = *(const v2f*)&As[wave * 16 + lm][k0];           // ds_load_b64
      #pragma unroll
      for (int nt = 0; nt < 4; ++nt) {
        v2f b;
        b.x = Ws[k0 + 0][nt * 16 + lm];
        b.y = Ws[k0 + 1][nt * 16 + lm];
        acc[nt] = wmma4(a, b, acc[nt]);
      }
    }
    __syncthreads();
  }

  #pragma unroll
  for (int nt = 0; nt < 4; ++nt) {
    int col = nb + nt * 16 + lm;
    #pragma unroll
    for (int v = 0; v < 8; ++v) {
      int row = mb + wave * 16 + half * 8 + v;
      float x = acc[nt][v];
      if (act == 1)      x = fmaxf(x, 0.0f);
      else if (act == 2) x = sigmf(x);
      Co[(size_t)row * ldc + col] = x;
    }
  }
}

// ---------------------------------------------------------------------------
// Persistent WMMA LSTM: one block = 16 batch rows, loops T steps.
// z[16,2048] = x_t[16,Kin]@Wx + h[16,512]@Wh + b, gates per reference:
//   i,f,o -> tanh ; g -> sigmoid ; c = f*c + i*g ; h = o*sigmoid(c)
// h lives in LDS (needed across waves for the K-dim), c lives in registers
// (lane->element mapping is static across steps).  rev=1 walks time
// backwards == flip -> lstm -> flip of the reference.
// Block: 256 thr = 8 waves; wave w owns cols {g*512 + w*64 .. +64} of each
// gate g, so the gate nonlinearity is wave-local.  Wx/Wh elements are each
// read exactly once per block per step (disjoint columns per wave) and are
// L2-resident, so they stream straight from global.
// ---------------------------------------------------------------------------
__global__ __launch_bounds__(256)
void k_lstm(const float* __restrict__ X, int Kin,
            const float* __restrict__ Wx,    // [Kin, 2048]
            const float* __restrict__ Wh,    // [512, 2048]
            const float* __restrict__ bias,  // [2048]
            float* __restrict__ Hout, int ldo, int co,
            int rev)
{
  __shared__ float hs[16][514];   // even stride: 8B-aligned v2f frags

  const int tid  = threadIdx.x;
  const int lane = tid & 31;
  const int wave = tid >> 5;     // 0..7
  const int half = lane >> 4;
  const int lm   = lane & 15;
  const int b0   = blockIdx.x * 16;

  for (int idx = tid; idx < 16 * 512; idx += 256)
    hs[idx >> 9][idx & 511] = 0.0f;
  __syncthreads();

  float bv[4][4];
  #pragma unroll
  for (int g = 0; g < 4; ++g)
    #pragma unroll
    for (int nt = 0; nt < 4; ++nt)
      bv[g][nt] = bias[g * 512 + wave * 64 + nt * 16 + lm];

  float creg[4][8];              // c state for this lane's (row,col) elems
  #pragma unroll
  for (int nt = 0; nt < 4; ++nt)
    #pragma unroll
    for (int v = 0; v < 8; ++v) creg[nt][v] = 0.0f;

  #pragma unroll 1
  for (int t = 0; t < TT; ++t) {
    const int tt = rev ? (TT - 1 - t) : t;
    const size_t xrow = ((size_t)(b0 + lm) * TT + tt) * (size_t)Kin;
    const v2f* xp = (const v2f*)(X + xrow);          // 8B-aligned

    // prefetch next timestep's x row (global_prefetch_b8)
    if (t + 1 < TT) {
      const int tn = rev ? (tt - 1) : (tt + 1);
      __builtin_prefetch(X + ((size_t)(b0 + lm) * TT + tn) * (size_t)Kin, 0, 0);
    }

    v8f acc[4][4];
    #pragma unroll
    for (int g = 0; g < 4; ++g)
      #pragma unroll
      for (int nt = 0; nt < 4; ++nt) {
        float b = bv[g][nt];
        v8f a0 = {b, b, b, b, b, b, b, b};
        acc[g][nt] = a0;
      }

    // ---- x_t @ Wx ----
    #pragma unroll 1
    for (int kt = 0; kt < Kin; kt += 4) {
      v2f a = xp[(kt >> 1) + half];                  // global_load_b64
      #pragma unroll
      for (int g = 0; g < 4; ++g) {
        #pragma unroll
        for (int nt = 0; nt < 4; ++nt) {
          int col = g * 512 + wave * 64 + nt * 16 + lm;
          v2f b;
          b.x = Wx[(size_t)(kt + half * 2 + 0) * G4H + col];
          b.y = Wx[(size_t)(kt + half * 2 + 1) * G4H + col];
          acc[g][nt] = wmma4(a, b, acc[g][nt]);
        }
      }
    }
    // ---- h @ Wh ----
    #pragma unroll 1
    for (int kt = 0; kt < HH; kt += 4) {
      v2f a = *(const v2f*)&hs[lm][kt + half * 2];   // ds_load_b64
      #pragma unroll
      for (int g = 0; g < 4; ++g) {
        #pragma unroll
        for (int nt = 0; nt < 4; ++nt) {
          int col = g * 512 + wave * 64 + nt * 16 + lm;
          v2f b;
          b.x = Wh[(size_t)(kt + half * 2 + 0) * G4H + col];
          b.y = Wh[(size_t)(kt + half * 2 + 1) * G4H + col];
          acc[g][nt] = wmma4(a, b, acc[g][nt]);
        }
      }
    }
    __syncthreads();   // everyone done reading hs

    // ---- gates + state update ----
    #pragma unroll
    for (int nt = 0; nt < 4; ++nt) {
      int j = wave * 64 + nt * 16 + lm;     // hidden index 0..511
      #pragma unroll
      for (int v = 0; v < 8; ++v) {
        int r = half * 8 + v;               // local batch row 0..15
        float iv = tanhf(acc[0][nt][v]);
        float fv = tanhf(acc[1][nt][v]);
        float gv = sigmf(acc[2][nt][v]);
        float ov = tanhf(acc[3][nt][v]);
        float c  = fv * creg[nt][v] + iv * gv;
        creg[nt][v] = c;
        float h  = ov * sigmf(c);
        hs[r][j] = h;
        Hout[((size_t)(b0 + r) * TT + tt) * (size_t)ldo + co + j] = h;
      }
    }
    __syncthreads();   // updates visible before next step reads hs
  }
}

// ---------------------------------------------------------------------------
// Attention core, one block per (batch, head), thread = query position.
// Online-softmax over K/V streamed through 64-row LDS chunks.
// q,k,v,o all laid out [B,T,NHEAD*HD] (head-major inner).  O may alias Q.
// ---------------------------------------------------------------------------
__global__ __launch_bounds__(256)
void k_attn(const float* __restrict__ Q, const float* __restrict__ Km,
            const float* __restrict__ V, float* __restrict__ O)
{
  __shared__ float ks[64][64];
  __shared__ float vs[64][64];

  const int b = blockIdx.x;
  const int h = blockIdx.y;
  const int t = threadIdx.x;                  // 0..255
  const size_t head_base = ((size_t)b * TT) * (size_t)HH + (size_t)h * HD;

  float q[HD];
  #pragma unroll
  for (int c = 0; c < HD; ++c)
    q[c] = Q[head_base + (size_t)t * HH + c] * 0.125f;   // 1/sqrt(64)

  float m = -3.0e38f, l = 0.0f;
  float o[HD];
  #pragma unroll
  for (int c = 0; c < HD; ++c) o[c] = 0.0f;

  #pragma unroll 1
  for (int ch = 0; ch < TT; ch += 64) {
    __syncthreads();
    for (int idx = t; idx < 64 * 64; idx += 256) {
      int r = idx >> 6, c = idx & 63;
      ks[r][c] = Km[head_base + (size_t)(ch + r) * HH + c];
      vs[r][c] = V [head_base + (size_t)(ch + r) * HH + c];
    }
    __syncthreads();
    #pragma unroll 1
    for (int s = 0; s < 64; ++s) {
      float sc = 0.0f;
      #pragma unroll
      for (int c = 0; c < HD; ++c) sc = __fmaf_rn(q[c], ks[s][c], sc);
      float mn   = fmaxf(m, sc);
      float corr = __expf(m - mn);
      float p    = __expf(sc - mn);
      l = l * corr + p;
      #pragma unroll
      for (int c = 0; c < HD; ++c) o[c] = o[c] * corr + p * vs[s][c];
      m = mn;
    }
  }
  float inv = 1.0f / l;
  #pragma unroll
  for (int c = 0; c < HD; ++c)
    O[head_base + (size_t)t * HH + c] = o[c] * inv;
}

// ---------------------------------------------------------------------------
// LayerNorm over last dim D, one block (128 thr) per row, in-place safe.
// ---------------------------------------------------------------------------
__global__ __launch_bounds__(128)
void k_ln(const float* __restrict__ Xi, float* __restrict__ Xo,
          const float* __restrict__ s, const float* __restrict__ b, int D)
{
  __shared__ float red[128];
  const int row = blockIdx.x;
  const float* xr = Xi + (size_t)row * D;

  float sum = 0.0f;
  for (int i = threadIdx.x; i < D; i += 128) sum += xr[i];
  red[threadIdx.x] = sum; __syncthreads();
  for (int o = 64; o > 0; o >>= 1) {
    if (threadIdx.x < o) red[threadIdx.x] += red[threadIdx.x + o];
    __syncthreads();
  }
  float mean = red[0] / (float)D;
  __syncthreads();

  float vsum = 0.0f;
  for (int i = threadIdx.x; i < D; i += 128) {
    float d = xr[i] - mean; vsum += d * d;
  }
  red[threadIdx.x] = vsum; __syncthreads();
  for (int o = 64; o > 0; o >>= 1) {
    if (threadIdx.x < o) red[threadIdx.x] += red[threadIdx.x + o];
    __syncthreads();
  }
  float rstd = rsqrtf(red[0] / (float)D + 1e-6f);
  __syncthreads();

  for (int i = threadIdx.x; i < D; i += 128)
    Xo[(size_t)row * D + i] = (xr[i] - mean) * rstd * s[i] + b[i];
}

// ---------------------------------------------------------------------------
// D = D * G + C  (elementwise, grid-stride)
// ---------------------------------------------------------------------------
__global__ void k_comb(float* __restrict__ D, const float* __restrict__ G,
                       const float* __restrict__ C, size_t n)
{
  size_t i = (size_t)blockIdx.x * blockDim.x + threadIdx.x;
  size_t st = (size_t)gridDim.x * blockDim.x;
  for (; i < n; i += st) D[i] = D[i] * G[i] + C[i];
}

// ---------------------------------------------------------------------------
// Pool + concat: E[b, 0:512]=mean_t Y, [512:1024]=max_t Y, [1024:1088]=other
// ---------------------------------------------------------------------------
__global__ void k_pool(const float* __restrict__ Y,
                       const float* __restrict__ other,
                       float* __restrict__ E)
{
  int idx = blockIdx.x * blockDim.x + threadIdx.x;
  if (idx >= BB * (2 * HH + FOTH)) return;
  int b = idx / (2 * HH + FOTH), col = idx % (2 * HH + FOTH);
  if (col < HH) {
    float s = 0.0f;
    for (int t = 0; t < TT; ++t) s += Y[((size_t)b * TT + t) * HH + col];
    E[idx] = s * (1.0f / (float)TT);
  } else if (col < 2 * HH) {
    int c = col - HH;
    float mx = -3.0e38f;
    for (int t = 0; t < TT; ++t) mx = fmaxf(mx, Y[((size_t)b * TT + t) * HH + c]);
    E[idx] = mx;
  } else {
    E[idx] = other[b * FOTH + (col - 2 * HH)];
  }
}

// ---------------------------------------------------------------------------
// Output head: out[b] = X[b,:]@w + b0     (128x128 -> 128)
// ---------------------------------------------------------------------------
__global__ void k_outhead(const float* __restrict__ X,
                          const float* __restrict__ w,
                          const float* __restrict__ b0,
                          float* __restrict__ out)
{
  int b = blockIdx.x * blockDim.x + threadIdx.x;
  if (b >= BB) return;
  float s = 0.0f;
  for (int k = 0; k < DD2; ++k) s += X[b * DD2 + k] * w[k];
  out[b] = s + b0[0];
}

// ---------------------------------------------------------------------------
// Host-side launch sequence
// ---------------------------------------------------------------------------
// input index map (setup_inputs dict flattened depth-first, insertion order)
enum {
  IN_CGM = 0, IN_OTHER,
  IN_DENSE_W, IN_DENSE_B, IN_LN0_S, IN_LN0_B,
  IN_LF_WX, IN_LF_WH, IN_LF_B,
  IN_LB_WX, IN_LB_WH, IN_LB_B,
  IN_LN1_S, IN_LN1_B,
  IN_LA_WX, IN_LA_WH, IN_LA_B,
  IN_LNA1_S, IN_LNA1_B,
  IN_WQ, IN_BQ, IN_WK, IN_BK, IN_WV, IN_BV, IN_WO, IN_BO,
  IN_GATE_W, IN_GATE_B, IN_LNA2_S, IN_LNA2_B,
  IN_D1_W, IN_D1_B, IN_LND1_S, IN_LND1_B,
  IN_D2_W, IN_D2_B, IN_LND2_S, IN_LND2_B,
  IN_OUT_W, IN_OUT_B
};

extern "C" void kernel_launch(void* const* d_in, const int* in_sizes, int n_in,
                              void* d_out, int out_size, void* d_ws, size_t ws_size,
                              hipStream_t stream) {
  (void)in_sizes; (void)n_in; (void)out_size; (void)ws_size;
  #define FIN(i) ((const float*)d_in[(i)])

  float* ws = (float*)d_ws;
  // workspace layout (floats)
  float* A  = ws;                             // [32768, 512]  x0 -> q -> attn o
  float* Bf = A  + (size_t)ROWS * HH;         // [32768,1024]  x1 -> k|v -> gate
  float* Cb = Bf + (size_t)ROWS * 2 * HH;     // [32768, 512]  h_a / skip2
  float* Db = Cb + (size_t)ROWS * HH;         // [32768, 512]  attn proj -> y
  float* E  = Db + (size_t)ROWS * HH;         // [128, 1088]
  float* E2 = E  + (size_t)BB * (2 * HH + FOTH); // [128, 256]
  float* E3 = E2 + (size_t)BB * DD1;          // [128, 128]
  float* Kbuf = Bf;                           // k in lower half of Bf
  float* Vbuf = Bf + (size_t)ROWS * HH;       // v in upper half of Bf

  // 1) x0 = LN0(cgm @ Win + bin)
  {
    dim3 g(HH / 64, ROWS / 64);
    k_gemm<<<g, 128, 0, stream>>>(FIN(IN_CGM), FCGM, FIN(IN_DENSE_W), HH,
                                  FIN(IN_DENSE_B), A, HH, ROWS, HH, FCGM, 0);
    k_ln<<<ROWS, 128, 0, stream>>>(A, A, FIN(IN_LN0_S), FIN(IN_LN0_B), HH);
  }

  // 2) bidirectional LSTM layer 0 -> Bf = [fwd | bwd], then LN1
  k_lstm<<<BB / 16, 256, 0, stream>>>(A, HH, FIN(IN_LF_WX), FIN(IN_LF_WH),
                                      FIN(IN_LF_B), Bf, 2 * HH, 0, 0);
  k_lstm<<<BB / 16, 256, 0, stream>>>(A, HH, FIN(IN_LB_WX), FIN(IN_LB_WH),
                                      FIN(IN_LB_B), Bf, 2 * HH, HH, 1);
  k_ln<<<ROWS, 128, 0, stream>>>(Bf, Bf, FIN(IN_LN1_S), FIN(IN_LN1_B), 2 * HH);

  // 3) LSTM layer a -> Cb, then LN_a1 (Cb = skip2)
  k_lstm<<<BB / 16, 256, 0, stream>>>(Bf, 2 * HH, FIN(IN_LA_WX), FIN(IN_LA_WH),
                                      FIN(IN_LA_B), Cb, HH, 0, 0);
  k_ln<<<ROWS, 128, 0, stream>>>(Cb, Cb, FIN(IN_LNA1_S), FIN(IN_LNA1_B), HH);

  // 4) attention: q->A, k->Kbuf, v->Vbuf, core writes o back into A
  {
    dim3 g(HH / 64, ROWS / 64);
    k_gemm<<<g, 128, 0, stream>>>(Cb, HH, FIN(IN_WQ), HH, FIN(IN_BQ), A,    HH, ROWS, HH, HH, 0);
    k_gemm<<<g, 128, 0, stream>>>(Cb, HH, FIN(IN_WK), HH, FIN(IN_BK), Kbuf, HH, ROWS, HH, HH, 0);
    k_gemm<<<g, 128, 0, stream>>>(Cb, HH, FIN(IN_WV), HH, FIN(IN_BV), Vbuf, HH, ROWS, HH, HH, 0);
    dim3 ga(BB, NHEAD);
    k_attn<<<ga, TT, 0, stream>>>(A, Kbuf, Vbuf, A);
    // out projection: Db = o @ Wo + bo
    k_gemm<<<g, 128, 0, stream>>>(A, HH, FIN(IN_WO), HH, FIN(IN_BO), Db, HH, ROWS, HH, HH, 0);
    // gate = sigmoid(skip2 @ gate_w + gate_b)  -> reuse Bf (k/v dead)
    k_gemm<<<g, 128, 0, stream>>>(Cb, HH, FIN(IN_GATE_W), HH, FIN(IN_GATE_B), Bf, HH, ROWS, HH, HH, 2);
  }

  // 5) y = LN_a2(attn*gate + skip2) in Db
  {
    size_t n = (size_t)ROWS * HH;
    k_comb<<<4096, 256, 0, stream>>>(Db, Bf, Cb, n);
    k_ln<<<ROWS, 128, 0, stream>>>(Db, Db, FIN(IN_LNA2_S), FIN(IN_LNA2_B), HH);
  }

  // 6) pooling + concat other -> E [128, 1088]
  {
    int tot = BB * (2 * HH + FOTH);
    k_pool<<<(tot + 255) / 256, 256, 0, stream>>>(Db, FIN(IN_OTHER), E);
  }

  // 7) dense head
  {
    dim3 g1(DD1 / 64, BB / 64);
    k_gemm<<<g1, 128, 0, stream>>>(E, 2 * HH + FOTH, FIN(IN_D1_W), DD1,
                                   FIN(IN_D1_B), E2, DD1, BB, DD1, 2 * HH + FOTH, 1);
    k_ln<<<BB, 128, 0, stream>>>(E2, E2, FIN(IN_LND1_S), FIN(IN_LND1_B), DD1);
    dim3 g2(DD2 / 64, BB / 64);
    k_gemm<<<g2, 128, 0, stream>>>(E2, DD1, FIN(IN_D2_W), DD2,
                                   FIN(IN_D2_B), E3, DD2, BB, DD2, DD1, 1);
    k_ln<<<BB, 128, 0, stream>>>(E3, E3, FIN(IN_LND2_S), FIN(IN_LND2_B), DD2);
    k_outhead<<<1, 128, 0, stream>>>(E3, FIN(IN_OUT_W), FIN(IN_OUT_B), (float*)d_out);
  }
  #undef FIN
}